// GCNConv_19825569038685
// MI455X (gfx1250) — compile-verified
//
#include <hip/hip_runtime.h>
#include <hip/hip_bf16.h>

typedef float v2f __attribute__((ext_vector_type(2)));
typedef float v8f __attribute__((ext_vector_type(8)));

#define F_DIM 256          // F_IN == F_OUT == 256
#define GEMM_CHUNK 25      // row-strips (of 16 rows) per wave

// ---------------------------------------------------------------------------
// Kernel 1: out[i] = bias[i % 256]   (base value for segment_sum + bias)
// ---------------------------------------------------------------------------
__global__ void gcn_init_out(const float* __restrict__ bias,
                             float* __restrict__ out, long long n) {
    long long i = (long long)blockIdx.x * blockDim.x + threadIdx.x;
    if (i < n) out[i] = bias[i & (F_DIM - 1)];
}

// ---------------------------------------------------------------------------
// Kernel 2: xw = x @ W via V_WMMA_F32_16X16X4_F32 (fp32 matrix pipe).
// Each wave owns one 16-wide column tile of W for the entire K range:
//   64 B-fragments x v2f = 128 VGPRs, loaded once, reused for 25 row strips.
// Per strip: 64 chained WMMAs, one b64 A-load each.
// ---------------------------------------------------------------------------
__global__ void __launch_bounds__(256)
gcn_gemm_xw(const float* __restrict__ x, const float* __restrict__ W,
            float* __restrict__ xw, int nStrips) {
    const int lane  = threadIdx.x & 31;
    const int wave  = threadIdx.x >> 5;
    const int gw    = blockIdx.x * 8 + wave;   // global wave id
    const int nt    = gw & 15;                 // column tile (0..15) -> cols nt*16..
    const int chunk = gw >> 4;                 // row-strip chunk
    const int colB  = lane & 15;               // N within tile / M-lane for A
    const int ksel  = (lane >> 4) << 1;        // lanes 0-15: K+0,K+1; 16-31: K+2,K+3

    // Preload B fragments for all K (W is 256 KB, L2-resident; read once per wave)
    v2f bfrag[64];
#pragma unroll
    for (int k = 0; k < 64; ++k) {
        const float* bp = W + (size_t)(k * 4 + ksel) * F_DIM + nt * 16 + colB;
        bfrag[k].x = bp[0];
        bfrag[k].y = bp[F_DIM];
    }

    for (int i = 0; i < GEMM_CHUNK; ++i) {
        const int strip = chunk * GEMM_CHUNK + i;
        if (strip >= nStrips) return;
        const int rowA = strip * 16 + (lane & 15);
        const float* ap = x + (size_t)rowA * F_DIM + ksel;

        v8f acc = {};
#pragma unroll
        for (int k = 0; k < 64; ++k) {
            v2f a = *(const v2f*)(ap + k * 4);   // 8B-aligned b64 load
            acc = __builtin_amdgcn_wmma_f32_16x16x4_f32(
                false, a, false, bfrag[k], (short)0, acc, false, false);
        }

        // D layout: VGPR r -> M = r (lanes 0-15) or M = 8+r (lanes 16-31)
        const int rowD = strip * 16 + ((lane >> 4) ? 8 : 0);
        float* op = xw + (size_t)rowD * F_DIM + nt * 16 + (lane & 15);
#pragma unroll
        for (int r = 0; r < 8; ++r) op[(size_t)r * F_DIM] = acc[r];
    }
}

// ---------------------------------------------------------------------------
// Kernel 3: scatter-add  out[dst[e]] += vals[e] * xw[src[e]]
// One wave per edge: lane handles 8 contiguous features (2x float4 gather,
// 8 fp32 HW atomics). out (102 MB) is L2-resident -> atomics resolve in L2.
// ---------------------------------------------------------------------------
__global__ void __launch_bounds__(256)
gcn_spmm_scatter(const float* __restrict__ xw, const int* __restrict__ esrc,
                 const int* __restrict__ edst, const float* __restrict__ evals,
                 float* __restrict__ out, int nEdges) {
    const int lane = threadIdx.x & 31;
    const int wave = threadIdx.x >> 5;
    const int e = blockIdx.x * 8 + wave;
    if (e >= nEdges) return;

    const int   s = esrc[e];
    const int   d = edst[e];
    const float v = evals[e];

    const float4* xr = (const float4*)(xw + (size_t)s * F_DIM);
    float* orow = out + (size_t)d * F_DIM + lane * 8;

    float4 p0 = xr[lane * 2];
    float4 p1 = xr[lane * 2 + 1];

    unsafeAtomicAdd(orow + 0, v * p0.x);
    unsafeAtomicAdd(orow + 1, v * p0.y);
    unsafeAtomicAdd(orow + 2, v * p0.z);
    unsafeAtomicAdd(orow + 3, v * p0.w);
    unsafeAtomicAdd(orow + 4, v * p1.x);
    unsafeAtomicAdd(orow + 5, v * p1.y);
    unsafeAtomicAdd(orow + 6, v * p1.z);
    unsafeAtomicAdd(orow + 7, v * p1.w);
}

// ---------------------------------------------------------------------------
extern "C" void kernel_launch(void* const* d_in, const int* in_sizes, int n_in,
                              void* d_out, int out_size, void* d_ws, size_t ws_size,
                              hipStream_t stream) {
    const float* x     = (const float*)d_in[0];
    const int*   esrc  = (const int*)  d_in[1];
    const int*   edst  = (const int*)  d_in[2];
    const float* evals = (const float*)d_in[3];
    const float* W     = (const float*)d_in[4];
    const float* bias  = (const float*)d_in[5];
    float* out = (float*)d_out;
    float* xw  = (float*)d_ws;            // [N_NODES, 256] fp32 scratch

    const int nNodes = in_sizes[0] / F_DIM;
    const int nEdges = in_sizes[1];
    const long long outElems = (long long)out_size;

    // 1) out = broadcast(bias)
    gcn_init_out<<<(int)((outElems + 255) / 256), 256, 0, stream>>>(bias, out, outElems);

    // 2) xw = x @ W  (WMMA fp32)
    const int nStrips = (nNodes + 15) / 16;                 // 6250
    const int nChunks = (nStrips + GEMM_CHUNK - 1) / GEMM_CHUNK;  // 250
    const int nWaves  = nChunks * 16;                       // 4000
    gcn_gemm_xw<<<(nWaves + 7) / 8, 256, 0, stream>>>(x, W, xw, nStrips);

    // 3) scatter: out[dst] += val * xw[src]
    gcn_spmm_scatter<<<(nEdges + 7) / 8, 256, 0, stream>>>(xw, esrc, edst, evals, out, nEdges);
}